// Permutation_12506944765981
// MI455X (gfx1250) — compile-verified
//
#include <hip/hip_runtime.h>
#include <stdint.h>

#define SUBS 8
#define LEADS 4096
#define TLEN 8192
#define BLK 1024  /* TLEN / SUBS floats per subsection (4 KiB) */

// Assume modern JAX default: jax_threefry_partitionable = True (default since ~0.4.36).
// Set to 0 to reproduce the legacy (original) threefry split/random_bits packing.
#define JAX_PARTITIONABLE 1

typedef float v4f __attribute__((ext_vector_type(4)));

// ---------------------------------------------------------------------------
// Threefry-2x32, 20 rounds — bit-exact with jax._src.prng.threefry2x32
// ---------------------------------------------------------------------------
__device__ __forceinline__ void threefry2x32(uint32_t k0, uint32_t k1,
                                             uint32_t c0, uint32_t c1,
                                             uint32_t& o0, uint32_t& o1) {
  const uint32_t ks2 = k0 ^ k1 ^ 0x1BD11BDAu;
  uint32_t x0 = c0 + k0;
  uint32_t x1 = c1 + k1;
  uint32_t ks[3] = {k1, ks2, k0};
  const uint32_t rotA[4] = {13u, 15u, 26u, 6u};
  const uint32_t rotB[4] = {17u, 29u, 16u, 24u};
#pragma unroll
  for (int i = 0; i < 5; ++i) {
    const uint32_t* rot = (i & 1) ? rotB : rotA;
#pragma unroll
    for (int r = 0; r < 4; ++r) {
      x0 += x1;
      x1 = (x1 << rot[r]) | (x1 >> (32u - rot[r]));
      x1 ^= x0;
    }
    x0 += ks[0];
    x1 += ks[1] + (uint32_t)(i + 1);
    uint32_t t = ks[0]; ks[0] = ks[1]; ks[1] = ks[2]; ks[2] = t;
  }
  o0 = x0; o1 = x1;
}

// ---------------------------------------------------------------------------
// Per-lead permutation: replicate
//   keys  = jax.random.split(jax.random.key(42), 4096)
//   perms = vmap(lambda k: jax.random.permutation(k, 8))(keys)
// permutation(k, 8) == one round of: key,sub = split(k);
//   bits = random_bits(sub, 32, (8,)); stable sort arange(8) by bits.
// ---------------------------------------------------------------------------
__global__ void perm_kernel(int* __restrict__ perms) {
  const uint32_t l = blockIdx.x * blockDim.x + threadIdx.x;
  if (l >= LEADS) return;

  uint32_t K0, K1, S0, S1;
  uint32_t bits[SUBS];

#if JAX_PARTITIONABLE
  // fold-like split: keys[l] = threefry((0,42), (0, l))
  threefry2x32(0u, 42u, 0u, l, K0, K1);
  // key,subkey = split(keys[l]) -> subkey = threefry(K, (0, 1))
  threefry2x32(K0, K1, 0u, 1u, S0, S1);
  // 32-bit draws: bits[j] = o0 ^ o1 of threefry(subkey, (0, j))
#pragma unroll
  for (uint32_t j = 0; j < SUBS; ++j) {
    uint32_t b0, b1;
    threefry2x32(S0, S1, 0u, j, b0, b1);
    bits[j] = b0 ^ b1;
  }
#else
  // Legacy packing: counts = iota(8192), halves (i, 4096+i), concat + reshape.
  {
    uint32_t a0, a1, d0, d1;
    if (l < 2048u) {
      threefry2x32(0u, 42u, 2u * l,      4096u + 2u * l,      a0, d0);
      threefry2x32(0u, 42u, 2u * l + 1u, 4097u + 2u * l,      a1, d1);
      K0 = a0; K1 = a1;
    } else {
      const uint32_t m = 2u * l - 4096u;
      threefry2x32(0u, 42u, m,      4096u + m, a0, d0);
      threefry2x32(0u, 42u, m + 1u, 4097u + m, a1, d1);
      K0 = d0; K1 = d1;
    }
  }
  // split(K): counts iota(4) halves -> subkey = (tf(K,(0,2)).o1, tf(K,(1,3)).o1)
  {
    uint32_t a0, b0, a1, b1;
    threefry2x32(K0, K1, 0u, 2u, a0, b0);
    threefry2x32(K0, K1, 1u, 3u, a1, b1);
    S0 = b0; S1 = b1;
  }
  // random_bits: counts iota(8) halves -> bits = [o0(j,4+j) x4, o1(j,4+j) x4]
#pragma unroll
  for (uint32_t j = 0; j < 4u; ++j) {
    uint32_t c, d;
    threefry2x32(S0, S1, j, 4u + j, c, d);
    bits[j] = c; bits[j + 4u] = d;
  }
#endif

  // Stable ascending sort of arange(8) keyed by bits (lax.sort_key_val).
  int perm[SUBS];
#pragma unroll
  for (int s = 0; s < SUBS; ++s) {
    int r = 0;
#pragma unroll
    for (int j = 0; j < SUBS; ++j)
      r += (bits[j] < bits[s]) || (bits[j] == bits[s] && j < s);
    perm[r] = s;
  }
#pragma unroll
  for (int s = 0; s < SUBS; ++s) perms[l * SUBS + s] = perm[s];
}

// ---------------------------------------------------------------------------
// Permuted row copy. CDNA5 path: async global->LDS->global B128 DMA
// (ASYNCcnt), no VGPR round trip. Fallback: nontemporal float4 copy.
// ---------------------------------------------------------------------------
#if defined(__gfx1250__) &&                                              \
    __has_builtin(__builtin_amdgcn_global_load_async_to_lds_b128) &&     \
    __has_builtin(__builtin_amdgcn_global_store_async_from_lds_b128)
#define USE_ASYNC_LDS 1
#else
#define USE_ASYNC_LDS 0
#endif

#if USE_ASYNC_LDS
// Builtins take pointer-to-(int __attribute__((vector_size(16)))) with
// addrspace(1) (global) / addrspace(3) (LDS) per the hipcc diagnostic.
typedef int v4i __attribute__((vector_size(16)));
typedef __attribute__((address_space(1))) v4i v4i_g;
typedef __attribute__((address_space(3))) v4i v4i_l;
// Flat->LDS: low 32 bits of the flat address are the LDS offset (aperture in
// the high bits), so truncation yields the addrspace(3) pointer value.
#define TO_GLOBAL(p) ((v4i_g*)(uintptr_t)(p))
#define TO_LDS(p) ((v4i_l*)(uint32_t)(uintptr_t)(p))

__device__ __forceinline__ void wait_async0() {
#if __has_builtin(__builtin_amdgcn_s_wait_asynccnt)
  __builtin_amdgcn_s_wait_asynccnt(0);
#else
  asm volatile("s_wait_asynccnt 0" ::: "memory");
#endif
}
#endif

__global__ void __launch_bounds__(256) permute_copy_kernel(
    const float* __restrict__ x, float* __restrict__ out,
    const int* __restrict__ perms) {
  const int l = blockIdx.x;     // one lead (row) per block
  const int tid = threadIdx.x;  // 256 threads; each moves 16 B per subsection
  const size_t rowBase = (size_t)l * TLEN;

  int p[SUBS];
#pragma unroll
  for (int s = 0; s < SUBS; ++s) p[s] = perms[l * SUBS + s];

#if USE_ASYNC_LDS
  __shared__ float lds[TLEN];  // 32 KiB staging (WGP has 320 KiB)
  // Each wave issues 8 async B128 loads (4 KiB in flight per wave), touching
  // only its own lanes' LDS slots -> a single per-wave s_wait_asynccnt
  // suffices; no workgroup barrier needed.
#pragma unroll
  for (int s = 0; s < SUBS; ++s) {
    const float* src = x + rowBase + (size_t)p[s] * BLK + (size_t)tid * 4;
    __builtin_amdgcn_global_load_async_to_lds_b128(
        TO_GLOBAL(src), TO_LDS(&lds[s * BLK + tid * 4]), 0, 0);
  }
  wait_async0();
#pragma unroll
  for (int s = 0; s < SUBS; ++s) {
    float* dst = out + rowBase + (size_t)s * BLK + (size_t)tid * 4;
    __builtin_amdgcn_global_store_async_from_lds_b128(
        TO_GLOBAL(dst), TO_LDS(&lds[s * BLK + tid * 4]), 0, 0);
  }
  wait_async0();  // drain stores (S_ENDPGM also waits idle)
#else
#pragma unroll
  for (int s = 0; s < SUBS; ++s) {
    const v4f v = __builtin_nontemporal_load(
        (const v4f*)(x + rowBase + (size_t)p[s] * BLK + (size_t)tid * 4));
    __builtin_nontemporal_store(
        v, (v4f*)(out + rowBase + (size_t)s * BLK + (size_t)tid * 4));
  }
#endif
}

// ---------------------------------------------------------------------------
extern "C" void kernel_launch(void* const* d_in, const int* in_sizes, int n_in,
                              void* d_out, int out_size, void* d_ws,
                              size_t ws_size, hipStream_t stream) {
  (void)in_sizes; (void)n_in; (void)out_size; (void)ws_size;
  const float* x = (const float*)d_in[0];
  float* out = (float*)d_out;
  int* perms = (int*)d_ws;  // 4096 * 8 * 4 B = 128 KiB scratch

  perm_kernel<<<(LEADS + 255) / 256, 256, 0, stream>>>(perms);
  permute_copy_kernel<<<LEADS, 256, 0, stream>>>(x, out, perms);
}